// LTM_7181185319107
// MI455X (gfx1250) — compile-verified
//
#include <hip/hip_runtime.h>
#include <hip/hip_bf16.h>

// LTM pipeline for MI455X (gfx1250, wave32).
// Box filters of the guided filter are executed on the WMMA units as banded
// matrix multiplies using V_WMMA_F32_16X16X4_F32.

typedef float v2f __attribute__((ext_vector_type(2)));
typedef float v8f __attribute__((ext_vector_type(8)));

#define FULL_H 2160
#define FULL_W 3840
#define DH 1080
#define DW 1920
#define RAD 8
#define INV17 (1.0f / 17.0f)
#define TILES_X (DW / 16)            // 120
#define TILES_Y ((DH + 15) / 16)     // 68 (last tile row-guarded)
#define NTILES (TILES_X * TILES_Y)   // 8160

__device__ __forceinline__ int refl(int i, int n) {
    // jnp.pad mode='reflect': -k -> k ; n-1+k -> n-1-k (RAD < n, single fold)
    i = (i < 0) ? -i : i;
    i = (i >= n) ? (2 * n - 2 - i) : i;
    return i;
}

// Banded box-filter weight: nonzero iff kk in [m, m+16]
__device__ __forceinline__ float bandw(int kk, int m) {
    return (kk >= m && kk <= m + 16) ? INV17 : 0.0f;
}

// ---------------- pass 1: luminance + log2 + exact 2x2 average ----------------
__global__ void down_kernel(const float* __restrict__ x, float* __restrict__ I) {
    int idx = blockIdx.x * blockDim.x + threadIdx.x;
    if (idx >= DW * DH) return;
    int xd = idx % DW, yd = idx / DW;
    float s = 0.0f;
#pragma unroll
    for (int dy = 0; dy < 2; ++dy)
#pragma unroll
        for (int dx = 0; dx < 2; ++dx) {
            size_t o = ((size_t)(2 * yd + dy) * FULL_W + (2 * xd + dx)) * 3;
            float lum = 0.2126f * x[o] + 0.7152f * x[o + 1] + 0.0722f * x[o + 2];
            s += log2f(fmaxf(lum, 1e-6f));
        }
    I[idx] = 0.25f * s;
}

// ------------- horizontal 17-tap box via WMMA: Out = In(16x32) x Band(32x16) -------------
// One wave per 16x16 output tile. Filters two planes sharing the band matrix:
// SQUARE=true  -> plane0 = s0, plane1 = s0^2   (mean_I / mean_II precursors)
// SQUARE=false -> plane0 = s0, plane1 = s1     (a / b)
template <bool SQUARE>
__global__ void hbox_kernel(const float* __restrict__ s0, const float* __restrict__ s1,
                            float* __restrict__ d0, float* __restrict__ d1) {
    int wave = blockIdx.x * (blockDim.x >> 5) + (threadIdx.x >> 5);
    if (wave >= NTILES) return;
    int tx = wave % TILES_X;
    int ty = wave / TILES_X;
    int lane = threadIdx.x & 31;
    int h = lane >> 4;   // lane half selects K pair
    int p = lane & 15;   // A: row (M) ; B: column (N)

    int rr = ty * 16 + p;
    if (rr > DH - 1) rr = DH - 1;  // overhang tile: load-safe, store-guarded
    const float* row0 = s0 + (size_t)rr * DW;
    const float* row1 = s1 + (size_t)rr * DW;

    v8f acc0 = {0, 0, 0, 0, 0, 0, 0, 0};
    v8f acc1 = {0, 0, 0, 0, 0, 0, 0, 0};
#pragma unroll
    for (int q = 0; q < 8; ++q) {
        int kk = 4 * q + 2 * h;              // K index of VGPR v is kk+v
        int c0 = tx * 16 - RAD + kk;         // input column for A element
        float a0 = row0[refl(c0, DW)];
        float a1 = row0[refl(c0 + 1, DW)];
        v2f av; av.x = a0; av.y = a1;
        v2f bv; bv.x = bandw(kk, p); bv.y = bandw(kk + 1, p);
        acc0 = __builtin_amdgcn_wmma_f32_16x16x4_f32(false, av, false, bv,
                                                     (short)0, acc0, false, false);
        v2f a2;
        if (SQUARE) { a2.x = a0 * a0; a2.y = a1 * a1; }
        else        { a2.x = row1[refl(c0, DW)]; a2.y = row1[refl(c0 + 1, DW)]; }
        acc1 = __builtin_amdgcn_wmma_f32_16x16x4_f32(false, a2, false, bv,
                                                     (short)0, acc1, false, false);
    }
    // D layout: element i of lane (16h+p) is Out[i+8h, p]
#pragma unroll
    for (int i = 0; i < 8; ++i) {
        int row = ty * 16 + i + 8 * h;
        if (row < DH) {
            size_t o = (size_t)row * DW + tx * 16 + p;
            d0[o] = acc0[i];
            d1[o] = acc1[i];
        }
    }
}

// ------------- vertical 17-tap box via WMMA: Out = Band(16x32) x In(32x16) -------------
// MODE 0: epilogue computes guided-filter coefficients a,b from mean_I/mean_II
// MODE 1: epilogue computes q = box(a)*I + box(b)
template <int MODE>
__global__ void vbox_kernel(const float* __restrict__ s0, const float* __restrict__ s1,
                            const float* __restrict__ img, float* __restrict__ d0,
                            float* __restrict__ d1) {
    int wave = blockIdx.x * (blockDim.x >> 5) + (threadIdx.x >> 5);
    if (wave >= NTILES) return;
    int tx = wave % TILES_X;
    int ty = wave / TILES_X;
    int lane = threadIdx.x & 31;
    int h = lane >> 4;
    int p = lane & 15;
    int col = tx * 16 + p;

    v8f acc0 = {0, 0, 0, 0, 0, 0, 0, 0};
    v8f acc1 = {0, 0, 0, 0, 0, 0, 0, 0};
#pragma unroll
    for (int q = 0; q < 8; ++q) {
        int kk = 4 * q + 2 * h;
        int r0 = refl(ty * 16 - RAD + kk, DH);
        int r1 = refl(ty * 16 - RAD + kk + 1, DH);
        v2f av; av.x = bandw(kk, p); av.y = bandw(kk + 1, p);   // A = band, M=p
        v2f b0; b0.x = s0[(size_t)r0 * DW + col]; b0.y = s0[(size_t)r1 * DW + col];
        v2f b1; b1.x = s1[(size_t)r0 * DW + col]; b1.y = s1[(size_t)r1 * DW + col];
        acc0 = __builtin_amdgcn_wmma_f32_16x16x4_f32(false, av, false, b0,
                                                     (short)0, acc0, false, false);
        acc1 = __builtin_amdgcn_wmma_f32_16x16x4_f32(false, av, false, b1,
                                                     (short)0, acc1, false, false);
    }
#pragma unroll
    for (int i = 0; i < 8; ++i) {
        int row = ty * 16 + i + 8 * h;
        if (row < DH) {
            size_t o = (size_t)row * DW + col;
            if (MODE == 0) {
                float mI = acc0[i], mII = acc1[i];
                float var = mII - mI * mI;
                float a = var / (var + 1e-3f);
                float b = mI - a * mI;
                d0[o] = a;
                d1[o] = b;
            } else {
                d0[o] = acc0[i] * img[o] + acc1[i];
            }
        }
    }
}

// ------------- final fused pass: upsample + tone map + rescale + clip -------------
__global__ void final_kernel(const float* __restrict__ x, const float* __restrict__ qimg,
                             float* __restrict__ out) {
    int idx = blockIdx.x * blockDim.x + threadIdx.x;
    if (idx >= FULL_W * FULL_H) return;
    int X = idx % FULL_W, Y = idx / FULL_W;
    size_t o = (size_t)idx * 3;
    float r = x[o], g = x[o + 1], b = x[o + 2];
    float lum = 0.2126f * r + 0.7152f * g + 0.0722f * b;
    float ylog = log2f(fmaxf(lum, 1e-6f));
    // bilinear upsample (half-pixel centers, align_corners=False)
    float ys = fmaxf((Y + 0.5f) * 0.5f - 0.5f, 0.0f);
    float xs = fmaxf((X + 0.5f) * 0.5f - 0.5f, 0.0f);
    float y0f = floorf(ys), x0f = floorf(xs);
    float wy = ys - y0f, wx = xs - x0f;
    int y0 = (int)y0f, x0 = (int)x0f;
    int y1 = min(y0 + 1, DH - 1), x1 = min(x0 + 1, DW - 1);
    float q00 = qimg[(size_t)y0 * DW + x0];
    float q01 = qimg[(size_t)y0 * DW + x1];
    float q10 = qimg[(size_t)y1 * DW + x0];
    float q11 = qimg[(size_t)y1 * DW + x1];
    float v0 = q00 * (1.0f - wy) + q10 * wy;   // vertical lerp first (matches ref)
    float v1 = q01 * (1.0f - wy) + q11 * wy;
    float ybase = v0 * (1.0f - wx) + v1 * wx;
    float ytm = 0.7f * ybase + (ylog - ybase);  // A_COMP*base + DETAIL_GAIN*detail
    float yout = exp2f(ytm);
    float scale = yout / (lum + 1e-4f);
    out[o]     = fminf(fmaxf(r * scale, 0.0f), 1.0f);
    out[o + 1] = fminf(fmaxf(g * scale, 0.0f), 1.0f);
    out[o + 2] = fminf(fmaxf(b * scale, 0.0f), 1.0f);
}

extern "C" void kernel_launch(void* const* d_in, const int* in_sizes, int n_in,
                              void* d_out, int out_size, void* d_ws, size_t ws_size,
                              hipStream_t stream) {
    const float* x = (const float*)d_in[0];
    float* out = (float*)d_out;
    float* wsf = (float*)d_ws;
    const size_t Nd = (size_t)DW * DH;
    float* I    = wsf;            // downsampled log-luminance
    float* t0   = wsf + Nd;       // hbox(I)  -> later hbox(a)
    float* t1   = wsf + 2 * Nd;   // hbox(I^2)-> later hbox(b)
    float* Abuf = wsf + 3 * Nd;   // guided 'a' ; later reused for q
    float* Bbuf = wsf + 4 * Nd;   // guided 'b'
    float* Q    = Abuf;           // q plane (Abuf not read in that pass)

    const int tblocks = (NTILES + 7) / 8;  // 8 waves (tiles) per 256-thread block

    down_kernel<<<(DW * DH + 255) / 256, 256, 0, stream>>>(x, I);
    hbox_kernel<true><<<tblocks, 256, 0, stream>>>(I, I, t0, t1);
    vbox_kernel<0><<<tblocks, 256, 0, stream>>>(t0, t1, nullptr, Abuf, Bbuf);
    hbox_kernel<false><<<tblocks, 256, 0, stream>>>(Abuf, Bbuf, t0, t1);
    vbox_kernel<1><<<tblocks, 256, 0, stream>>>(t0, t1, I, Q, nullptr);
    final_kernel<<<(FULL_W * FULL_H + 255) / 256, 256, 0, stream>>>(x, Q, out);
}